// WCELoss_83863531422179
// MI455X (gfx1250) — compile-verified
//
#include <hip/hip_runtime.h>
#include <stdint.h>

#define N_ 8
#define C_ 11
#define H_ 512
#define W_ 512
#define HW_ (H_ * W_)            // 262144 = 1<<18
#define NHW_ (N_ * HW_)          // 2097152
#define NC_ (N_ * C_)            // 88
#define IGNORE_ 255
#define TILE_PX 1024
#define NTILES (NHW_ / TILE_PX)  // 2048
#define SCALE (1.0f / (float)((long long)N_ * C_ * H_ * W_))  // 1/23068672

// ---------------- CDNA5 async global->LDS path ----------------
#if defined(__has_builtin)
#if __has_builtin(__builtin_amdgcn_global_load_async_to_lds_b128)
#define HAVE_ASYNC_LDS 1
#endif
#if __has_builtin(__builtin_amdgcn_s_wait_asynccnt)
#define HAVE_WAIT_ASYNC 1
#endif
#endif

typedef int v4i_ __attribute__((vector_size(4 * sizeof(int))));
typedef __attribute__((address_space(1))) v4i_ gv4i;
typedef __attribute__((address_space(3))) v4i_ lv4i;

__device__ __forceinline__ void async_copy_b128(const void* gsrc, void* ldst) {
#if defined(HAVE_ASYNC_LDS)
  // flat->AS1 / flat->AS3 via integer cast (low 32 bits of LDS flat addr = LDS offset)
  gv4i* gp = (gv4i*)(uintptr_t)gsrc;
  lv4i* lp = (lv4i*)(uint32_t)(uintptr_t)ldst;
  __builtin_amdgcn_global_load_async_to_lds_b128(gp, lp, 0, 0);
#else
  *(float4*)ldst = *(const float4*)gsrc;
#endif
}

__device__ __forceinline__ void wait_async0() {
#if defined(HAVE_WAIT_ASYNC)
  __builtin_amdgcn_s_wait_asynccnt(0);
#else
  asm volatile("s_wait_asynccnt 0" ::: "memory");
#endif
  asm volatile("" ::: "memory");  // keep LDS reads after the wait
}

// ---------------- k1: zero the count accumulators ----------------
__global__ void k_zero(float* __restrict__ counts) {
  if (threadIdx.x < NC_) counts[threadIdx.x] = 0.0f;
}

// ---------------- k2: per-(n,c) valid-pixel counts ----------------
__global__ void __launch_bounds__(256) k_count(const long long* __restrict__ labels,
                                               float* __restrict__ counts) {
  __shared__ float scnt[NC_];
  int t = threadIdx.x;
  if (t < NC_) scnt[t] = 0.0f;
  __syncthreads();
  long long stride = (long long)gridDim.x * blockDim.x;
  for (long long p = (long long)blockIdx.x * blockDim.x + t; p < NHW_; p += stride) {
    int lab = (int)labels[p];
    if (lab != IGNORE_ && (unsigned)lab < (unsigned)C_) {
      int n = (int)(p >> 18);  // p / HW_
      atomicAdd(&scnt[n * C_ + lab], 1.0f);
    }
  }
  __syncthreads();
  if (t < NC_ && scnt[t] != 0.0f) atomicAdd(&counts[t], scnt[t]);
}

// ---------------- k3: freq -> lower median -> weights ----------------
__global__ void k_weights(const float* __restrict__ counts, float* __restrict__ wout) {
  __shared__ float freq[NC_];
  __shared__ float colsum[C_];
  __shared__ float median_s;
  int t = threadIdx.x;
  if (t < C_) {
    float s = 0.0f;
    for (int n = 0; n < N_; ++n) s += counts[n * C_ + t];
    colsum[t] = s;
  }
  __syncthreads();
  if (t < NC_) freq[t] = counts[t] / colsum[t % C_];
  __syncthreads();
  if (t < NC_) {
    // unique rank (stable): lower median = element with rank (NC_-1)/2 == 43
    float v = freq[t];
    int rank = 0;
    for (int j = 0; j < NC_; ++j) {
      float u = freq[j];
      rank += (u < v) || (u == v && j < t);
    }
    if (rank == (NC_ - 1) / 2) median_s = v;
  }
  __syncthreads();
  if (t < NC_) {
    float f = freq[t];
    float m = median_s;
    float fe = (f == 0.0f) ? m : f;
    wout[t] = m / fe;
  }
}

// ---------------- k4: weighted NLL over one 1024-pixel tile ----------------
__global__ void __launch_bounds__(256) k_loss(const float* __restrict__ logits,
                                              const long long* __restrict__ labels,
                                              const float* __restrict__ wtab,
                                              float* __restrict__ partials) {
  __shared__ float lbuf[C_ * TILE_PX];  // 44 KB staged logits
  __shared__ float wsm[NC_];
  __shared__ float wavesum[8];
  const int tid = threadIdx.x;
  const long long pixBase = (long long)blockIdx.x * TILE_PX;
  const int n = (int)(pixBase >> 18);            // pixBase / HW_
  const long long inPlane = pixBase & (HW_ - 1); // tile never crosses an image
  const float* base = logits + (size_t)n * C_ * HW_ + inPlane;

  // stage: each lane async-copies its own 16B per class (self-consumed -> wave-local wait only)
#pragma unroll
  for (int c = 0; c < C_; ++c)
    async_copy_b128(base + (size_t)c * HW_ + tid * 4, &lbuf[c * TILE_PX + tid * 4]);

  if (tid < NC_) wsm[tid] = wtab[tid];
  __syncthreads();   // publish wsm
  wait_async0();     // our own staged logits are now in LDS

  float acc = 0.0f;
  const long long* lb = labels + pixBase + tid * 4;
#pragma unroll
  for (int p = 0; p < 4; ++p) {
    const int q = tid * 4 + p;
    float r[C_];
#pragma unroll
    for (int c = 0; c < C_; ++c) r[c] = lbuf[c * TILE_PX + q];
    float m = r[0];
#pragma unroll
    for (int c = 1; c < C_; ++c) m = fmaxf(m, r[c]);
    float s = 0.0f;
#pragma unroll
    for (int c = 0; c < C_; ++c) s += __expf(r[c] - m);
    int lab = (int)lb[p];
    bool valid = (lab != IGNORE_);
    int cl = valid ? lab : 0;
    if ((unsigned)cl >= (unsigned)C_) { valid = false; cl = 0; }
    float xl = lbuf[cl * TILE_PX + q];      // LDS gather avoids register-array spill
    float nll = (m + __logf(s)) - xl;
    float wgt = wsm[n * C_ + cl];
    acc += valid ? wgt * nll : 0.0f;
  }

  // wave32 reduce, then cross-wave
#pragma unroll
  for (int off = 16; off > 0; off >>= 1) acc += __shfl_down(acc, off, 32);
  if ((tid & 31) == 0) wavesum[tid >> 5] = acc;
  __syncthreads();
  if (tid == 0) {
    float s = 0.0f;
#pragma unroll
    for (int i = 0; i < 8; ++i) s += wavesum[i];
    partials[blockIdx.x] = s;  // no atomics: deterministic
  }
}

// ---------------- k5: fixed-order final reduction ----------------
__global__ void __launch_bounds__(256) k_final(const float* __restrict__ partials,
                                               float* __restrict__ out) {
  __shared__ float red[256];
  int t = threadIdx.x;
  float s = 0.0f;
  for (int i = t; i < NTILES; i += 256) s += partials[i];
  red[t] = s;
  __syncthreads();
  for (int off = 128; off > 0; off >>= 1) {
    if (t < off) red[t] += red[t + off];
    __syncthreads();
  }
  if (t == 0) out[0] = red[0] * SCALE;
}

extern "C" void kernel_launch(void* const* d_in, const int* in_sizes, int n_in,
                              void* d_out, int out_size, void* d_ws, size_t ws_size,
                              hipStream_t stream) {
  const float* logits = (const float*)d_in[0];
  const long long* labels = (const long long*)d_in[1];
  float* ws = (float*)d_ws;
  float* counts = ws;             // [0, 88)
  float* wtab = ws + NC_;         // [88, 176)
  float* partials = ws + 2 * NC_; // [176, 176+2048)
  float* out = (float*)d_out;

  k_zero<<<1, 128, 0, stream>>>(counts);
  k_count<<<1024, 256, 0, stream>>>(labels, counts);
  k_weights<<<1, 128, 0, stream>>>(counts, wtab);
  k_loss<<<NTILES, 256, 0, stream>>>(logits, labels, wtab, partials);
  k_final<<<1, 256, 0, stream>>>(partials, out);
}